// BERT_ner_40226663695003
// MI455X (gfx1250) — compile-verified
//
#include <hip/hip_runtime.h>
#include <hip/hip_bf16.h>

// CRF loss (BERT NER) for gfx1250.
// Normalizer scan recast as per-step f32 GEMM in exp space -> V_WMMA_F32_16X16X4_F32.
//   T=512, B=2048, K=48.  Workgroup = 96 threads (3 waves), owns 16 batch rows;
//   wave w owns j-tile w (16 output tags). E = exp(transitions) lives in regs.
// Latency plan: logits tile for step t+1 double-buffered in regs (issued at top of
// step t), t+4 tile prefetched into cache (global_prefetch_b8); wave-local S-tile
// roundtrip fenced with s_wait_dscnt instead of a workgroup barrier.

#define T_LEN 512
#define B_SZ  2048
#define K_TAG 48
#define SROW  52            // padded LDS row stride in floats (bank-conflict avoidance)
#define PFD   4             // cache-prefetch distance (steps)

typedef float v2f __attribute__((ext_vector_type(2)));
typedef float v4f __attribute__((ext_vector_type(4)));
typedef float v8f __attribute__((ext_vector_type(8)));

__device__ __forceinline__ v4f max4(v4f a, v4f b) {
  v4f r;
  r.x = fmaxf(a.x, b.x); r.y = fmaxf(a.y, b.y);
  r.z = fmaxf(a.z, b.z); r.w = fmaxf(a.w, b.w);
  return r;
}

// wave-local LDS fence: S-tile is written and read by the same wave only.
__device__ __forceinline__ void lds_fence_wave() {
  asm volatile("s_wait_dscnt 0x0" ::: "memory");
}

__global__ __launch_bounds__(96) void crf_normalizer(
    const float* __restrict__ logits,            // (T, B, K)
    const float* __restrict__ masks,             // (T, B)
    const float* __restrict__ start_transitions, // (K)
    const float* __restrict__ end_transitions,   // (K)
    const float* __restrict__ transitions,       // (K, K)
    float* __restrict__ out_norm)                // (B)
{
  __shared__ __align__(16) float alphaBuf[2][16 * SROW];
  __shared__ __align__(16) float Sbuf[16 * SROW];

  const int tid   = threadIdx.x;
  const int w     = tid >> 5;       // wave id 0..2 == j-tile
  const int lane  = tid & 31;
  const int col16 = lane & 15;      // N within tile / batch-row id in row phases
  const int half  = lane >> 4;      // 0 or 1 (WMMA K-half)
  const int b0    = blockIdx.x * 16;

  // ---- constant B matrix: E = exp(transitions), WMMA 4x16 f32 B layout ----
  // (lanes 0-15: K=4c,4c+1 ; lanes 16-31: K=4c+2,4c+3 ; N = w*16 + col16)
  v2f Bm[12];
  {
    const int n0 = w * 16 + col16;
#pragma unroll
    for (int c = 0; c < 12; ++c) {
      const int i0 = 4 * c + 2 * half;
      Bm[c].x = __expf(transitions[(i0 + 0) * K_TAG + n0]);
      Bm[c].y = __expf(transitions[(i0 + 1) * K_TAG + n0]);
    }
  }

  // end_transitions for this wave's 16 columns (added every step: faithful to ref)
  float et[16];
#pragma unroll
  for (int c = 0; c < 16; ++c) et[c] = end_transitions[w * 16 + c];

  // per-lane base pointer for this lane's logit slice (row b0+col16, cols w*16..)
  const float* lgBase = logits + (size_t)(b0 + col16) * K_TAG + w * 16;
  const size_t lgStep = (size_t)B_SZ * K_TAG;

  // ---- t = 0 : alpha0 = start_transitions + logits[0] ----
  if (lane < 16) {
    float* aw = &alphaBuf[0][col16 * SROW + w * 16];
#pragma unroll
    for (int c = 0; c < 16; c += 4) {
      v4f l = *(const v4f*)(lgBase + c);
      v4f o;
      o.x = start_transitions[w * 16 + c + 0] + l.x;
      o.y = start_transitions[w * 16 + c + 1] + l.y;
      o.z = start_transitions[w * 16 + c + 2] + l.z;
      o.w = start_transitions[w * 16 + c + 3] + l.w;
      *(v4f*)(aw + c) = o;
    }
  }

  // ---- preload logit tile + mask for t = 1 into registers ----
  v4f LgC[4] = {};
  float mkC = 1.f;
  if (lane < 16) {
    const float* p = lgBase + 1 * lgStep;
#pragma unroll
    for (int c = 0; c < 4; ++c) LgC[c] = *(const v4f*)(p + 4 * c);
    mkC = masks[(size_t)1 * B_SZ + b0 + col16];
  }
  __syncthreads();

  // ---- scan t = 1 .. T-1 ----
  for (int t = 1; t < T_LEN; ++t) {
    const int pb = (t - 1) & 1;
    const int cb = t & 1;

    // Issue next-step loads + far prefetch first: a full step of work hides them.
    v4f LgN[4] = {};
    float mkN = 1.f;
    const bool haveNext = (t + 1 < T_LEN);
    if (lane < 16) {
      if (haveNext) {
        const float* p = lgBase + (size_t)(t + 1) * lgStep;
#pragma unroll
        for (int c = 0; c < 4; ++c) LgN[c] = *(const v4f*)(p + 4 * c);
        mkN = masks[(size_t)(t + 1) * B_SZ + b0 + col16];
      }
      if (t + PFD < T_LEN)
        __builtin_prefetch(lgBase + (size_t)(t + PFD) * lgStep, 0, 1);
    }

    // A-prep: every lane reads full alpha row (row = col16) from LDS
    v4f row[12];
    {
      const float* ar = &alphaBuf[pb][col16 * SROW];
#pragma unroll
      for (int c = 0; c < 12; ++c) row[c] = *(const v4f*)(ar + 4 * c);
    }
    v4f mx = row[0];
#pragma unroll
    for (int c = 1; c < 12; ++c) mx = max4(mx, row[c]);
    const float m = fmaxf(fmaxf(mx.x, mx.y), fmaxf(mx.z, mx.w));

    // A regs: p = exp(alpha - m) in WMMA 16x4 f32 A layout
    v2f A[12];
#pragma unroll
    for (int c = 0; c < 12; ++c) {
      const float a0 = half ? row[c].z : row[c].x;
      const float a1 = half ? row[c].w : row[c].y;
      A[c].x = __expf(a0 - m);
      A[c].y = __expf(a1 - m);
    }

    // 12 x V_WMMA_F32_16X16X4_F32, 3 rotating accumulators (chain length 4)
    v8f acc0{}, acc1{}, acc2{};
#pragma unroll
    for (int c = 0; c < 12; c += 3) {
      acc0 = __builtin_amdgcn_wmma_f32_16x16x4_f32(false, A[c + 0], false, Bm[c + 0],
                                                   (short)0, acc0, false, false);
      acc1 = __builtin_amdgcn_wmma_f32_16x16x4_f32(false, A[c + 1], false, Bm[c + 1],
                                                   (short)0, acc1, false, false);
      acc2 = __builtin_amdgcn_wmma_f32_16x16x4_f32(false, A[c + 2], false, Bm[c + 2],
                                                   (short)0, acc2, false, false);
    }
    v8f acc = acc0 + acc1 + acc2;

    // D layout (lane holds rows v / v+8 at col = w*16+col16) -> LDS S tile.
    // Written and read by this wave only -> wave-local dscnt fence, no barrier.
#pragma unroll
    for (int v = 0; v < 8; ++v) {
      const int rr = half ? (v + 8) : v;
      Sbuf[rr * SROW + w * 16 + col16] = acc[v];
    }
    lds_fence_wave();

    // Compose: alpha'[r][j] = (mask ? m + log(S) + logit : alpha_old) + end_trans
    if (lane < 16) {
      const int r = col16;
      const bool on = mkC > 0.f;
      const float* sr = &Sbuf[r * SROW + w * 16];
      const float* ao = &alphaBuf[pb][r * SROW + w * 16];
      float* aw = &alphaBuf[cb][r * SROW + w * 16];
#pragma unroll
      for (int c = 0; c < 16; c += 4) {
        v4f sv = *(const v4f*)(sr + c);
        v4f lv = LgC[c / 4];
        v4f ov = *(const v4f*)(ao + c);
        v4f o;
        o.x = (on ? (m + __logf(sv.x) + lv.x) : ov.x) + et[c + 0];
        o.y = (on ? (m + __logf(sv.y) + lv.y) : ov.y) + et[c + 1];
        o.z = (on ? (m + __logf(sv.z) + lv.z) : ov.z) + et[c + 2];
        o.w = (on ? (m + __logf(sv.w) + lv.w) : ov.w) + et[c + 3];
        *(v4f*)(aw + c) = o;
      }
    }
    __syncthreads();

    // rotate next-step registers into current
    if (haveNext) {
#pragma unroll
      for (int c = 0; c < 4; ++c) LgC[c] = LgN[c];
      mkC = mkN;
    }
  }

  // ---- final: norm[b] = logsumexp_j alpha[b][j] ----
  if (w == 0 && lane < 16) {
    const float* ar = &alphaBuf[(T_LEN - 1) & 1][col16 * SROW];
    v4f rows[12];
    rows[0] = *(const v4f*)(ar);
    v4f mxv = rows[0];
#pragma unroll
    for (int c = 1; c < 12; ++c) {
      rows[c] = *(const v4f*)(ar + 4 * c);
      mxv = max4(mxv, rows[c]);
    }
    const float m = fmaxf(fmaxf(mxv.x, mxv.y), fmaxf(mxv.z, mxv.w));
    float s = 0.f;
#pragma unroll
    for (int c = 0; c < 12; ++c) {
      s += __expf(rows[c].x - m) + __expf(rows[c].y - m) +
           __expf(rows[c].z - m) + __expf(rows[c].w - m);
    }
    out_norm[b0 + col16] = m + __logf(s);
  }
}

// ---- gold-path score: one thread per batch element, 8-step gather batches ----
__global__ __launch_bounds__(256) void crf_score(
    const float* __restrict__ logits, const float* __restrict__ masks,
    const int* __restrict__ labels, const float* __restrict__ start_transitions,
    const float* __restrict__ end_transitions, const float* __restrict__ transitions,
    float* __restrict__ out_score)
{
  const int b = blockIdx.x * blockDim.x + threadIdx.x;
  if (b >= B_SZ) return;
  int prev = labels[b];
  float s = start_transitions[prev] + logits[(size_t)b * K_TAG + prev];
  float cnt = masks[b];

  int t = 1;
  for (; t + 8 <= T_LEN; t += 8) {
    int   lt[8];
    float mk[8];
#pragma unroll
    for (int u = 0; u < 8; ++u) {
      lt[u] = labels[(size_t)(t + u) * B_SZ + b];
      mk[u] = masks[(size_t)(t + u) * B_SZ + b];
    }
#pragma unroll
    for (int u = 0; u < 8; ++u) {
      s += mk[u] * (transitions[prev * K_TAG + lt[u]] +
                    logits[((size_t)(t + u) * B_SZ + b) * K_TAG + lt[u]]);
      prev = lt[u];
      cnt += mk[u];
    }
  }
  for (; t < T_LEN; ++t) {
    const int lt = labels[(size_t)t * B_SZ + b];
    const float mk = masks[(size_t)t * B_SZ + b];
    s += mk * (transitions[prev * K_TAG + lt] +
               logits[((size_t)t * B_SZ + b) * K_TAG + lt]);
    prev = lt;
    cnt += mk;
  }
  const int send = (int)cnt - 1;
  const int last = labels[(size_t)send * B_SZ + b];
  out_score[b] = s + end_transitions[last];
}

// ---- masks.sum() partial reduction ----
__global__ __launch_bounds__(256) void crf_mask_sum(
    const float* __restrict__ masks, float* __restrict__ part)
{
  __shared__ float sm[256];
  const int tid = threadIdx.x;
  float s = 0.f;
  for (size_t i = (size_t)blockIdx.x * 256 + tid; i < (size_t)T_LEN * B_SZ;
       i += (size_t)256 * gridDim.x)
    s += masks[i];
  sm[tid] = s;
  __syncthreads();
  for (int k = 128; k > 0; k >>= 1) {
    if (tid < k) sm[tid] += sm[tid + k];
    __syncthreads();
  }
  if (tid == 0) part[blockIdx.x] = sm[0];
}

// ---- final scalar: sum(score - norm) / masks.sum() ----
__global__ __launch_bounds__(256) void crf_finalize(
    const float* __restrict__ score, const float* __restrict__ norm,
    const float* __restrict__ part, int nparts, float* __restrict__ out)
{
  __shared__ float sm[256];
  const int tid = threadIdx.x;
  float s = 0.f;
  for (int i = tid; i < B_SZ; i += 256) s += score[i] - norm[i];
  sm[tid] = s;
  __syncthreads();
  for (int k = 128; k > 0; k >>= 1) {
    if (tid < k) sm[tid] += sm[tid + k];
    __syncthreads();
  }
  const float diff = sm[0];
  __syncthreads();
  float p = 0.f;
  for (int i = tid; i < nparts; i += 256) p += part[i];
  sm[tid] = p;
  __syncthreads();
  for (int k = 128; k > 0; k >>= 1) {
    if (tid < k) sm[tid] += sm[tid + k];
    __syncthreads();
  }
  if (tid == 0) out[0] = diff / sm[0];
}

extern "C" void kernel_launch(void* const* d_in, const int* in_sizes, int n_in,
                              void* d_out, int out_size, void* d_ws, size_t ws_size,
                              hipStream_t stream) {
  (void)in_sizes; (void)n_in; (void)out_size; (void)ws_size;
  const float* logits = (const float*)d_in[0];
  const float* masks  = (const float*)d_in[1];
  const int*   labels = (const int*)d_in[2];
  const float* st     = (const float*)d_in[3];
  const float* et     = (const float*)d_in[4];
  const float* trans  = (const float*)d_in[5];

  float* ws_norm  = (float*)d_ws;          // [B_SZ]
  float* ws_score = ws_norm + B_SZ;        // [B_SZ]
  float* ws_part  = ws_score + B_SZ;       // [128]

  crf_normalizer<<<B_SZ / 16, 96, 0, stream>>>(logits, masks, st, et, trans, ws_norm);
  crf_score<<<(B_SZ + 255) / 256, 256, 0, stream>>>(logits, masks, labels, st, et, trans, ws_score);
  crf_mask_sum<<<128, 256, 0, stream>>>(masks, ws_part);
  crf_finalize<<<1, 256, 0, stream>>>(ws_score, ws_norm, ws_part, 128, (float*)d_out);
}